// GCLLayer_68478958567603
// MI455X (gfx1250) — compile-verified
//
#include <hip/hip_runtime.h>
#include <hip/hip_bf16.h>

// ---------------------------------------------------------------------------
// GCN layer:  support = features @ W^T + bias ; out = scatter_add(support[cols]*vals, rows)
//   features: [N,128] f32   weight: [128,128] f32   bias: [128] f32
//   edges:    vals f32 [E], rows i32 [E], cols i32 [E]
// ---------------------------------------------------------------------------

typedef float v2f __attribute__((ext_vector_type(2)));
typedef float v8f __attribute__((ext_vector_type(8)));

#define D_DIM 128  // D_IN == D_OUT == 128

// --------------------------- zero the output -------------------------------
__global__ void gcl_zero4(float* __restrict__ p, size_t n4) {
    size_t i = (size_t)blockIdx.x * blockDim.x + threadIdx.x;
    if (i < n4) ((float4*)p)[i] = make_float4(0.f, 0.f, 0.f, 0.f);
}

// --------------------- dense GEMM via V_WMMA_F32_16X16X4_F32 ---------------
// One wave (32 lanes) computes one 16x16 tile of `support`.
// tile id -> (mtile, ntile); 8 ntiles cover the 128 output columns.
// A layout (16x4 f32, ISA 7.12.2): lane = 16*half + M, VGPR j = K = 2*half + j
// B layout (4x16 f32):             lane = 16*half + N, VGPR j = K = 2*half + j
//   B[k][n] = weight[n][k]  (support = feat @ W^T), so each lane streams row n of W.
// C layout (16x16 f32, 8 VGPRs):   lanes 0-15: N=lane, M=i ; lanes 16-31: N=lane-16, M=8+i
__global__ __launch_bounds__(128) void gcl_gemm_wmma(
    const float* __restrict__ feat,
    const float* __restrict__ weight,
    const float* __restrict__ bias,
    float* __restrict__ support,
    int nRows)
{
    const int lane   = threadIdx.x & 31;
    const int wave   = threadIdx.x >> 5;
    const int tile   = blockIdx.x * 4 + wave;
    const int mtile  = tile >> 3;          // row tile (16 rows)
    const int ntile  = tile & 7;           // col tile (16 cols of 128)
    const int m0     = mtile * 16;
    if (m0 >= nRows) return;               // wave-uniform

    const int half   = lane >> 4;          // selects K pair {0,1} vs {2,3}
    const int l16    = lane & 15;
    const int koff   = half * 2;

    // clamp A row for (non-occurring with N=100000) ragged last tile
    const int arow = (m0 + l16 < nRows) ? (m0 + l16) : (nRows - 1);
    const float* aPtr = feat   + (size_t)arow * D_DIM + koff;
    const float* bPtr = weight + (size_t)(ntile * 16 + l16) * D_DIM + koff;

    v8f c = {};
#pragma unroll
    for (int k0 = 0; k0 < D_DIM; k0 += 4) {
        v2f a = *(const v2f*)(aPtr + k0);
        v2f b = *(const v2f*)(bPtr + k0);
        // 8 args: (neg_a, A, neg_b, B, c_mod, C, reuse_a, reuse_b)
        c = __builtin_amdgcn_wmma_f32_16x16x4_f32(
                false, a, false, b, (short)0, c, false, false);
    }

    const int   ncol = ntile * 16 + l16;
    const float bv   = bias[ncol];
#pragma unroll
    for (int i = 0; i < 8; ++i) {
        int row = m0 + half * 8 + i;
        if (row < nRows)
            support[(size_t)row * D_DIM + ncol] = c[i] + bv;
    }
}

// ------------------ COO SpMM: gather row, scale, scatter-add ---------------
// One wave per edge: 32 lanes x float4 = 128-wide row. support/output live in
// the 192MB L2, so gathers and the non-returning global_atomic_add_f32 stream
// are L2-resident.
__global__ __launch_bounds__(256) void gcl_spmm_scatter(
    const float* __restrict__ support,
    const float* __restrict__ vals,
    const int*   __restrict__ rows,
    const int*   __restrict__ cols,
    float* __restrict__ out,
    int nEdges)
{
    const int lane = threadIdx.x & 31;
    const int wave = threadIdx.x >> 5;
    const int e    = blockIdx.x * 8 + wave;
    if (e >= nEdges) return;               // wave-uniform

    const int   col = cols[e];
    const int   row = rows[e];
    const float v   = vals[e];

    const float4 s = ((const float4*)(support + (size_t)col * D_DIM))[lane];
    float* dst = out + (size_t)row * D_DIM + lane * 4;
    atomicAdd(dst + 0, s.x * v);   // lowers to global_atomic_add_f32 (no return)
    atomicAdd(dst + 1, s.y * v);
    atomicAdd(dst + 2, s.z * v);
    atomicAdd(dst + 3, s.w * v);
}

// ---------------------------------------------------------------------------
extern "C" void kernel_launch(void* const* d_in, const int* in_sizes, int n_in,
                              void* d_out, int out_size, void* d_ws, size_t ws_size,
                              hipStream_t stream) {
    const float* feat   = (const float*)d_in[0];
    const float* weight = (const float*)d_in[1];
    const float* bias   = (const float*)d_in[2];
    const float* evals  = (const float*)d_in[3];
    const int*   erows  = (const int*)d_in[4];
    const int*   ecols  = (const int*)d_in[5];
    float*       out    = (float*)d_out;

    const int nNodes = in_sizes[0] / D_DIM;
    const int nEdges = in_sizes[3];
    float* support = (float*)d_ws;         // [nNodes, 128] f32 scratch

    // 1) zero output (we scatter-add into it)
    {
        size_t n4 = (size_t)nNodes * D_DIM / 4;
        int blocks = (int)((n4 + 255) / 256);
        gcl_zero4<<<blocks, 256, 0, stream>>>(out, n4);
    }

    // 2) support = feat @ W^T + bias   (WMMA fp32)
    {
        int mtiles = (nNodes + 15) / 16;
        int tiles  = mtiles * 8;           // 8 col tiles
        int blocks = (tiles + 3) / 4;      // 4 waves per block
        gcl_gemm_wmma<<<blocks, 128, 0, stream>>>(feat, weight, bias, support, nNodes);
    }

    // 3) out[rows] += support[cols] * vals   (one wave per edge)
    {
        int blocks = (nEdges + 7) / 8;     // 8 waves per block
        gcl_spmm_scatter<<<blocks, 256, 0, stream>>>(support, evals, erows, ecols, out, nEdges);
    }
}